// ExpertTower_67783173865959
// MI455X (gfx1250) — compile-verified
//
#include <hip/hip_runtime.h>
#include <hip/hip_bf16.h>

#define HDIM 128

typedef float v2f __attribute__((ext_vector_type(2)));
typedef float v8f __attribute__((ext_vector_type(8)));

// One wave (32 lanes) per edge: gather 128-float source row as coalesced
// float4s (512B), scatter-add into dest row with 4 non-returning f32 atomics
// per lane. x (51MB) lives in L2, atomics execute at L2.
__global__ __launch_bounds__(256) void scatter_add_kernel(
    const float* __restrict__ x, const int* __restrict__ src,
    const int* __restrict__ dst, float* __restrict__ agg, int nedges)
{
    int e = blockIdx.x * 8 + (int)(threadIdx.x >> 5);
    if (e >= nedges) return;
    int lane = threadIdx.x & 31;
    int s = src[e];
    int d = dst[e];
    const float4* xrow = (const float4*)(x + (size_t)s * HDIM);
    float4 v = xrow[lane];
    float* arow = agg + (size_t)d * HDIM + (size_t)lane * 4;
    atomicAdd(arow + 0, v.x);
    atomicAdd(arow + 1, v.y);
    atomicAdd(arow + 2, v.z);
    atomicAdd(arow + 3, v.w);
}

// out = act(A1 @ W1 + A2 @ W2 + bias), all dims [nrows x 128] @ [128 x 128].
// One wave per 16x16 output tile using V_WMMA_F32_16X16X4_F32 (exact fp32).
// Block = 256 threads = 8 waves = 16 rows x 128 cols.
__global__ __launch_bounds__(256) void gemm2_bias_act_kernel(
    const float* __restrict__ A1, const float* __restrict__ W1,
    const float* __restrict__ A2, const float* __restrict__ W2,
    const float* __restrict__ bias, float* __restrict__ out,
    int do_relu)
{
    const int lane = (int)(threadIdx.x & 31);
    const int wave = (int)(threadIdx.x >> 5);
    const int row0 = (int)blockIdx.x * 16;
    const int col0 = wave * 16;
    const int m    = lane & 15;         // A row within tile; also B/C col (n)
    const int kb   = (lane >> 4) << 1;  // K sub-offset within 4-wide step: 0 or 2
    const int rsel = (lane >> 4) << 3;  // C/D row offset: 0 or 8

    const float* a1p = A1 + (size_t)(row0 + m) * HDIM;
    const float* a2p = A2 + (size_t)(row0 + m) * HDIM;
    const float* w1p = W1 + col0 + m;   // W[k][n] row-major, stride HDIM in k
    const float* w2p = W2 + col0 + m;

    // Init accumulator with bias (same for every row of the tile).
    const float bv = bias[col0 + m];
    v8f acc;
#pragma unroll
    for (int j = 0; j < 8; ++j) acc[j] = bv;

#pragma unroll 4
    for (int k = 0; k < HDIM; k += 4) {
        // A tiles: contiguous float2 per lane (8B-aligned since k+kb is even)
        v2f a1 = *(const v2f*)(a1p + k + kb);
        v2f b1;
        b1.x = w1p[(size_t)(k + kb)     * HDIM];
        b1.y = w1p[(size_t)(k + kb + 1) * HDIM];
        acc = __builtin_amdgcn_wmma_f32_16x16x4_f32(
            false, a1, false, b1, (short)0, acc, false, false);

        v2f a2 = *(const v2f*)(a2p + k + kb);
        v2f b2;
        b2.x = w2p[(size_t)(k + kb)     * HDIM];
        b2.y = w2p[(size_t)(k + kb + 1) * HDIM];
        acc = __builtin_amdgcn_wmma_f32_16x16x4_f32(
            false, a2, false, b2, (short)0, acc, false, false);
    }

    float* orow = out + (size_t)(row0 + rsel) * HDIM + col0 + m;
#pragma unroll
    for (int j = 0; j < 8; ++j) {
        float v = acc[j];
        if (do_relu) v = fmaxf(v, 0.0f);
        orow[(size_t)j * HDIM] = v;
    }
}

extern "C" void kernel_launch(void* const* d_in, const int* in_sizes, int n_in,
                              void* d_out, int out_size, void* d_ws, size_t ws_size,
                              hipStream_t stream) {
    (void)n_in; (void)out_size; (void)ws_size;

    const float* x     = (const float*)d_in[0];
    const int*   edges = (const int*)d_in[1];   // [2, E] flat
    const float* Wrel0 = (const float*)d_in[2];
    const float* Wrt0  = (const float*)d_in[3];
    const float* b0    = (const float*)d_in[4];
    const float* Wrel1 = (const float*)d_in[5];
    const float* Wrt1  = (const float*)d_in[6];
    const float* b1    = (const float*)d_in[7];
    const float* Wrel2 = (const float*)d_in[8];
    const float* Wrt2  = (const float*)d_in[9];
    const float* b2    = (const float*)d_in[10];

    const int N = in_sizes[0] / HDIM;           // 100000
    const int E = in_sizes[1] / 2;              // 1600000
    const int* src = edges;
    const int* dst = edges + E;

    const size_t feat = (size_t)N * HDIM;       // floats per node-feature buffer
    float* agg = (float*)d_ws;
    float* hA  = agg + feat;
    float* hB  = hA + feat;
    float* y   = (float*)d_out;

    const dim3 blk(256);
    const dim3 gridScatter((unsigned)((E + 7) / 8));
    const dim3 gridGemm((unsigned)((N + 15) / 16));

    // ---- layer 0: hA = relu(agg(x) @ Wrel0 + x @ Wrt0 + b0) ----
    hipMemsetAsync(agg, 0, feat * sizeof(float), stream);
    scatter_add_kernel<<<gridScatter, blk, 0, stream>>>(x, src, dst, agg, E);
    gemm2_bias_act_kernel<<<gridGemm, blk, 0, stream>>>(agg, Wrel0, x, Wrt0, b0, hA, 1);

    // ---- layer 1: hB = relu(agg(hA) @ Wrel1 + hA @ Wrt1 + b1) ----
    hipMemsetAsync(agg, 0, feat * sizeof(float), stream);
    scatter_add_kernel<<<gridScatter, blk, 0, stream>>>(hA, src, dst, agg, E);
    gemm2_bias_act_kernel<<<gridGemm, blk, 0, stream>>>(agg, Wrel1, hA, Wrt1, b1, hB, 1);

    // ---- layer 2: out = agg(hB) @ Wrel2 + hB @ Wrt2 + b2 ----
    hipMemsetAsync(agg, 0, feat * sizeof(float), stream);
    scatter_add_kernel<<<gridScatter, blk, 0, stream>>>(hB, src, dst, agg, E);
    gemm2_bias_act_kernel<<<gridGemm, blk, 0, stream>>>(agg, Wrel2, hB, Wrt2, b2, y, 0);
}